// BeliefPropagation_41515153883659
// MI455X (gfx1250) — compile-verified
//
#include <hip/hip_runtime.h>
#include <math.h>

// Problem constants (fixed by the reference).
#define NA 1048576
#define NB 65536
#define RR 32
#define NAR (NA * RR)   // 33,554,432 elements of ta
#define NBR (NB * RR)   //  2,097,152 elements of tb

typedef __attribute__((ext_vector_type(2))) float v2f;
typedef __attribute__((ext_vector_type(8))) float v8f;

// ---- order-preserving float <-> uint encoding (for atomicMax on floats) ----
__device__ __forceinline__ unsigned f2ord(float f) {
    unsigned u = __float_as_uint(f);
    return (u & 0x80000000u) ? ~u : (u | 0x80000000u);
}
__device__ __forceinline__ float ord2f(unsigned s) {
    unsigned u = (s & 0x80000000u) ? (s & 0x7FFFFFFFu) : ~s;
    return __uint_as_float(u);
}

// ---- block max reduction (blockDim.x == 256, wave32) ----
__device__ __forceinline__ float blockMax256(float v) {
    for (int off = 16; off > 0; off >>= 1)
        v = fmaxf(v, __shfl_xor(v, off, 32));
    __shared__ float smax[8];
    const int lane = threadIdx.x & 31, wid = threadIdx.x >> 5;
    if (lane == 0) smax[wid] = v;
    __syncthreads();
    if (wid == 0) {
        v = (lane < 8) ? smax[lane] : -INFINITY;
        for (int off = 4; off > 0; off >>= 1)
            v = fmaxf(v, __shfl_xor(v, off, 32));
    }
    return v;  // valid in thread 0
}

// ---- K1: init workspace ----
__global__ void bp_init(unsigned* __restrict__ maxu, float* __restrict__ sums,
                        unsigned* __restrict__ scal, float* __restrict__ fsum) {
    const int i = blockIdx.x * blockDim.x + threadIdx.x;
    if (i < NBR) { maxu[i] = 0u; sums[i] = 0.0f; }
    if (i < 2)   { scal[i] = 0u; fsum[i] = 0.0f; }
}

// ---- K2: scatter-max of theta_a into per-segment max table (float4 stream) ----
__global__ void bp_scatter_max(const float* __restrict__ ta,
                               const int* __restrict__ idx,
                               unsigned* __restrict__ maxu) {
    const int i = blockIdx.x * blockDim.x + threadIdx.x;  // float4 index, < NAR/4
    const int row = i >> 3;                               // 8 float4 per row of 32
    const int seg = idx[row];                             // uniform per 8 lanes
    const int j = (seg << 5) | ((i & 7) << 2);
    const float4 x = reinterpret_cast<const float4*>(ta)[i];  // global_load_b128
    atomicMax(&maxu[j + 0], f2ord(x.x));
    atomicMax(&maxu[j + 1], f2ord(x.y));
    atomicMax(&maxu[j + 2], f2ord(x.z));
    atomicMax(&maxu[j + 3], f2ord(x.w));
}

// ---- K3: scatter-add exp(x - segmax), gather theta_b, write ta, reduce max(ta) ----
__global__ void bp_scatter_add_gather(const float* __restrict__ ta,
                                      const float* __restrict__ tb,
                                      const int* __restrict__ idx,
                                      const unsigned* __restrict__ maxu,
                                      float* __restrict__ sums,
                                      float* __restrict__ out_ta,
                                      unsigned* __restrict__ ta_maxu) {
    const int i = blockIdx.x * blockDim.x + threadIdx.x;  // float4 index, < NAR/4
    const int row = i >> 3;
    const int seg = idx[row];
    const int j4 = (seg << 3) | (i & 7);                  // float4/uint4 index into [NB,32]
    const float4 x = reinterpret_cast<const float4*>(ta)[i];
    const uint4  mu = reinterpret_cast<const uint4*>(maxu)[j4];
    float4 sh;
    sh.x = x.x - ord2f(mu.x); sh.y = x.y - ord2f(mu.y);
    sh.z = x.z - ord2f(mu.z); sh.w = x.w - ord2f(mu.w);
    if (sh.x != sh.x) sh.x = 0.0f;
    if (sh.y != sh.y) sh.y = 0.0f;
    if (sh.z != sh.z) sh.z = 0.0f;
    if (sh.w != sh.w) sh.w = 0.0f;
    float* sp = sums + ((size_t)j4 << 2);
    atomicAdd(sp + 0, expf(sh.x));                        // global_atomic_add_f32 (L2-resident)
    atomicAdd(sp + 1, expf(sh.y));
    atomicAdd(sp + 2, expf(sh.z));
    atomicAdd(sp + 3, expf(sh.w));
    const float4 b = reinterpret_cast<const float4*>(tb)[j4];   // m_ba gather (L2-resident)
    float4 t;
    t.x = x.x + b.x; t.y = x.y + b.y; t.z = x.z + b.z; t.w = x.w + b.w;
    reinterpret_cast<float4*>(out_ta)[i] = t;             // global_store_b128
    const float tm = fmaxf(fmaxf(t.x, t.y), fmaxf(t.z, t.w));
    const float bm = blockMax256(tm);
    if (threadIdx.x == 0) atomicMax(ta_maxu, f2ord(bm));
}

// ---- K4: tb_un = theta_b + log(sums) + segmax; reduce max(tb) ----
__global__ void bp_make_tb(const float* __restrict__ tb,
                           const float* __restrict__ sums,
                           const unsigned* __restrict__ maxu,
                           float* __restrict__ out_tb,
                           unsigned* __restrict__ tb_maxu) {
    const int i = blockIdx.x * blockDim.x + threadIdx.x;  // float4 index, < NBR/4
    const float4 b = reinterpret_cast<const float4*>(tb)[i];
    const float4 s = reinterpret_cast<const float4*>(sums)[i];
    const uint4  mu = reinterpret_cast<const uint4*>(maxu)[i];
    float4 t;
    t.x = b.x + logf(s.x) + ord2f(mu.x);
    t.y = b.y + logf(s.y) + ord2f(mu.y);
    t.z = b.z + logf(s.z) + ord2f(mu.z);
    t.w = b.w + logf(s.w) + ord2f(mu.w);
    reinterpret_cast<float4*>(out_tb)[i] = t;
    const float tm = fmaxf(fmaxf(t.x, t.y), fmaxf(t.z, t.w));
    const float bm = blockMax256(tm);
    if (threadIdx.x == 0) atomicMax(tb_maxu, f2ord(bm));
}

// ---- K5: global sum of exp(x - gmax) using the matrix pipe ----
// Each wave folds 128-element chunks (one b128 load per lane) into a 16x16 f32
// tile via two V_WMMA_F32_16X16X4_F32 ops with B == ones:
//   D[m,n] = sum_k A[m,k] + C[m,n]  =>  column 0 of the tile = running total,
// held by lane 0 (rows 0..7) and lane 16 (rows 8..15).
__global__ void bp_sumexp_wmma(const float* __restrict__ x, int nchunks,
                               const unsigned* __restrict__ gmax_ord,
                               float* __restrict__ out_sum) {
    const float gmax = ord2f(*gmax_ord);
    const int lane   = threadIdx.x & 31;
    const int wave   = (blockIdx.x * blockDim.x + threadIdx.x) >> 5;
    const int nwaves = (gridDim.x * blockDim.x) >> 5;

    v8f acc = {};
    v2f ones; ones.x = 1.0f; ones.y = 1.0f;
    const float4* xv = reinterpret_cast<const float4*>(x);

    for (int ch = wave; ch < nchunks; ch += nwaves) {     // uniform per wave: EXEC all-1s
        const float4* p = xv + (size_t)ch * 32 + lane;
        __builtin_prefetch(p + (size_t)nwaves * 32, 0, 0); // global_prefetch_b8
        const float4 v = *p;                               // global_load_b128
        v2f a0, a1;
        a0.x = expf(v.x - gmax); a0.y = expf(v.y - gmax);
        a1.x = expf(v.z - gmax); a1.y = expf(v.w - gmax);
        acc = __builtin_amdgcn_wmma_f32_16x16x4_f32(
            false, a0, false, ones, (short)0, acc, false, false);
        acc = __builtin_amdgcn_wmma_f32_16x16x4_f32(
            false, a1, false, ones, (short)0, acc, false, false);
    }
    float s = acc[0] + acc[1] + acc[2] + acc[3] + acc[4] + acc[5] + acc[6] + acc[7];
    float tot = __shfl(s, 0, 32) + __shfl(s, 16, 32);     // column 0 = chunk totals
    if (lane == 0) atomicAdd(out_sum, tot);
}

// ---- K6: subtract per-tensor logsumexp (float4 stream) ----
__global__ void bp_normalize(float* __restrict__ out,
                             const unsigned* __restrict__ scal,
                             const float* __restrict__ fsum) {
    const int i = blockIdx.x * blockDim.x + threadIdx.x;  // float4 index
    const bool isA = i < (NAR / 4);
    const float lse = isA ? (logf(fsum[0]) + ord2f(scal[0]))
                          : (logf(fsum[1]) + ord2f(scal[1]));
    float4 v = reinterpret_cast<float4*>(out)[i];
    v.x -= lse; v.y -= lse; v.z -= lse; v.w -= lse;
    reinterpret_cast<float4*>(out)[i] = v;
}

extern "C" void kernel_launch(void* const* d_in, const int* in_sizes, int n_in,
                              void* d_out, int out_size, void* d_ws, size_t ws_size,
                              hipStream_t stream) {
    (void)in_sizes; (void)n_in; (void)out_size; (void)ws_size;
    const float* theta_a = (const float*)d_in[0];
    const float* theta_b = (const float*)d_in[1];
    const int*   idx_map = (const int*)d_in[2];
    float* out = (float*)d_out;
    float* out_ta = out;                 // [NA*R]
    float* out_tb = out + (size_t)NAR;   // [NB*R]

    // Workspace layout: maxu[NBR] u32 | sums[NBR] f32 | scal[2] u32 | fsum[2] f32
    unsigned* maxu = (unsigned*)d_ws;
    float*    sums = (float*)((char*)d_ws + (size_t)NBR * 4);
    unsigned* scal = (unsigned*)((char*)d_ws + (size_t)NBR * 8);
    float*    fsum = (float*)((char*)d_ws + (size_t)NBR * 8 + 8);

    const int BLK = 256;

    bp_init<<<NBR / BLK, BLK, 0, stream>>>(maxu, sums, scal, fsum);

    bp_scatter_max<<<NAR / 4 / BLK, BLK, 0, stream>>>(theta_a, idx_map, maxu);

    bp_scatter_add_gather<<<NAR / 4 / BLK, BLK, 0, stream>>>(
        theta_a, theta_b, idx_map, maxu, sums, out_ta, &scal[0]);

    bp_make_tb<<<NBR / 4 / BLK, BLK, 0, stream>>>(theta_b, sums, maxu, out_tb, &scal[1]);

    // Sum-exp passes (WMMA reduction). Chunk = 128 elements per wave-iteration.
    bp_sumexp_wmma<<<1024, BLK, 0, stream>>>(out_ta, NAR / 128, &scal[0], &fsum[0]);
    bp_sumexp_wmma<<<256,  BLK, 0, stream>>>(out_tb, NBR / 128, &scal[1], &fsum[1]);

    bp_normalize<<<(NAR + NBR) / 4 / BLK, BLK, 0, stream>>>(out, scal, fsum);
}